// GraphSage_25898652795240
// MI455X (gfx1250) — compile-verified
//
#include <hip/hip_runtime.h>
#include <hip/hip_bf16.h>
#include <math.h>

// GraphSAGE 2-layer inference for MI455X (gfx1250, wave32).
// Aggregation is L2-resident (x: 25.6MB, msg: 25.6MB << 192MB L2) -> f32 atomics.
// GEMMs (4.9 GFLOP total) done in exact fp32 with V_WMMA_F32_16X16X4_F32.

#define NN      50000
#define NE      800000
#define IN_CH   128
#define HID_CH  128
#define OUT_CH  64

typedef float v2f __attribute__((ext_vector_type(2)));
typedef float v8f __attribute__((ext_vector_type(8)));

__global__ void zero_f32(float* __restrict__ p, int n) {
    int i = blockIdx.x * blockDim.x + threadIdx.x;
    if (i < n) p[i] = 0.0f;
}

// deg[dst] += 1 per edge
__global__ void degree_kernel(const int* __restrict__ ei, float* __restrict__ deg) {
    int e = blockIdx.x * blockDim.x + threadIdx.x;
    if (e < NE) atomicAdd(&deg[ei[NE + e]], 1.0f);
}

// One thread per (edge, 4-channel group): float4 gather from feat[src], 4 f32 atomics into msg[dst].
// Works for both layers (both have 128 input channels).
__global__ void scatter_kernel(const float* __restrict__ feat, const int* __restrict__ ei,
                               float* __restrict__ msg) {
    long g = (long)blockIdx.x * blockDim.x + threadIdx.x;
    if (g >= (long)NE * 32) return;
    int e  = (int)(g >> 5);
    int c4 = ((int)g & 31) << 2;
    int src = ei[e];
    int dst = ei[NE + e];
    const float4 v = *(const float4*)(feat + (size_t)src * 128 + c4);
    float* o = msg + (size_t)dst * 128 + c4;
    atomicAdd(o + 0, v.x);
    atomicAdd(o + 1, v.y);
    atomicAdd(o + 2, v.z);
    atomicAdd(o + 3, v.w);
}

// msg[i] /= max(deg[node],1)   (128 channels per node)
__global__ void mean_kernel(float* __restrict__ msg, const float* __restrict__ deg) {
    int i = blockIdx.x * blockDim.x + threadIdx.x;
    if (i < NN * 128) {
        float d = fmaxf(deg[i >> 7], 1.0f);
        msg[i] = msg[i] / d;
    }
}

// One wave computes a 16x16 output tile of  A1*W1^T + A2*W2^T  (K=128) with
// 64x V_WMMA_F32_16X16X4_F32 (exact fp32, f32 accumulate).
__device__ __forceinline__ v8f dual_gemm_tile(const float* __restrict__ A1,
                                              const float* __restrict__ A2,
                                              const float* __restrict__ W1,
                                              const float* __restrict__ W2,
                                              int mBase, int nBase, int K) {
    const int lane = threadIdx.x & 31;
    const int r    = lane & 15;          // row for A, col for B (same index)
    const int kk   = (lane >> 4) << 1;   // 0 or 2: K sub-offset for this half-wave
    const float* a1 = A1 + (size_t)(mBase + r) * K + kk;
    const float* a2 = A2 + (size_t)(mBase + r) * K + kk;
    const float* b1 = W1 + (size_t)(nBase + r) * K + kk;  // W stored (n,k) row-major = B^T rows
    const float* b2 = W2 + (size_t)(nBase + r) * K + kk;
    v8f acc = {};
#pragma unroll 8
    for (int k0 = 0; k0 < K; k0 += 4) {
        v2f av = *(const v2f*)(a1 + k0);
        v2f bv = *(const v2f*)(b1 + k0);
        acc = __builtin_amdgcn_wmma_f32_16x16x4_f32(false, av, false, bv,
                                                    (short)0, acc, false, false);
        v2f aw = *(const v2f*)(a2 + k0);
        v2f bw = *(const v2f*)(b2 + k0);
        acc = __builtin_amdgcn_wmma_f32_16x16x4_f32(false, aw, false, bw,
                                                    (short)0, acc, false, false);
    }
    return acc;
}

// Layer 1: h = relu(agg @ W1l^T + x @ W1r^T + b1).  block = 256 (8 waves = 8 n-tiles of 16).
__global__ void gemm1_relu(const float* __restrict__ agg, const float* __restrict__ x,
                           const float* __restrict__ W1l, const float* __restrict__ W1r,
                           const float* __restrict__ b1, float* __restrict__ h) {
    const int mBase = blockIdx.x * 16;
    const int nBase = (threadIdx.x >> 5) * 16;
    v8f acc = dual_gemm_tile(agg, x, W1l, W1r, mBase, nBase, IN_CH);
    const int lane   = threadIdx.x & 31;
    const int col    = lane & 15;
    const int rowOff = (lane >> 4) << 3;
    const float bias = b1[nBase + col];
#pragma unroll
    for (int v = 0; v < 8; ++v) {
        float y = fmaxf(acc[v] + bias, 0.0f);
        h[(size_t)(mBase + rowOff + v) * HID_CH + nBase + col] = y;
    }
}

// Layer 2 + log_softmax over 64 outputs.  block = 128 (4 waves = 4 n-tiles of 16).
__global__ void gemm2_logsoftmax(const float* __restrict__ agg, const float* __restrict__ hin,
                                 const float* __restrict__ W2l, const float* __restrict__ W2r,
                                 const float* __restrict__ b2, float* __restrict__ out) {
    __shared__ float tile[16][OUT_CH];
    const int mBase = blockIdx.x * 16;
    const int nBase = (threadIdx.x >> 5) * 16;
    v8f acc = dual_gemm_tile(agg, hin, W2l, W2r, mBase, nBase, HID_CH);
    const int lane   = threadIdx.x & 31;
    const int col    = lane & 15;
    const int rowOff = (lane >> 4) << 3;
    const float bias = b2[nBase + col];
#pragma unroll
    for (int v = 0; v < 8; ++v)
        tile[rowOff + v][nBase + col] = acc[v] + bias;
    __syncthreads();
    const int r = threadIdx.x;
    if (r < 16) {
        float mx = -3.402823466e38f;
        for (int j = 0; j < OUT_CH; ++j) mx = fmaxf(mx, tile[r][j]);
        float s = 0.0f;
        for (int j = 0; j < OUT_CH; ++j) s += __expf(tile[r][j] - mx);
        const float lg = mx + __logf(s);
        for (int j = 0; j < OUT_CH; ++j)
            out[(size_t)(mBase + r) * OUT_CH + j] = tile[r][j] - lg;
    }
}

extern "C" void kernel_launch(void* const* d_in, const int* in_sizes, int n_in,
                              void* d_out, int out_size, void* d_ws, size_t ws_size,
                              hipStream_t stream) {
    (void)in_sizes; (void)n_in; (void)out_size; (void)ws_size;
    const float* x   = (const float*)d_in[0];
    const int*   ei  = (const int*)  d_in[1];   // edge_index [2, NE] (int32 on device)
    const float* W1l = (const float*)d_in[2];
    const float* b1  = (const float*)d_in[3];
    const float* W1r = (const float*)d_in[4];
    const float* W2l = (const float*)d_in[5];
    const float* b2  = (const float*)d_in[6];
    const float* W2r = (const float*)d_in[7];
    float* out = (float*)d_out;

    // workspace: msg [NN*128] | deg [NN] | h [NN*128]
    float* msg = (float*)d_ws;
    float* deg = msg + (size_t)NN * 128;
    float* h   = deg + NN;

    const int feat_total = NN * 128;
    const long sthreads  = (long)NE * 32;
    const int  sblocks   = (int)((sthreads + 255) / 256);

    // ---- layer 1 aggregation ----
    zero_f32<<<(feat_total + NN + 255) / 256, 256, 0, stream>>>(msg, feat_total + NN);
    degree_kernel<<<(NE + 255) / 256, 256, 0, stream>>>(ei, deg);
    scatter_kernel<<<sblocks, 256, 0, stream>>>(x, ei, msg);
    mean_kernel<<<(feat_total + 255) / 256, 256, 0, stream>>>(msg, deg);
    // ---- layer 1 dual-GEMM + relu ----
    gemm1_relu<<<NN / 16, 256, 0, stream>>>(msg, x, W1l, W1r, b1, h);
    // ---- layer 2 aggregation (reuse msg, deg unchanged) ----
    zero_f32<<<(feat_total + 255) / 256, 256, 0, stream>>>(msg, feat_total);
    scatter_kernel<<<sblocks, 256, 0, stream>>>(h, ei, msg);
    mean_kernel<<<(feat_total + 255) / 256, 256, 0, stream>>>(msg, deg);
    // ---- layer 2 dual-GEMM + log_softmax ----
    gemm2_logsoftmax<<<NN / 16, 128, 0, stream>>>(msg, h, W2l, W2r, b2, out);
}